// CoevolutionDeletionHead_36558761624136
// MI455X (gfx1250) — compile-verified
//
#include <hip/hip_runtime.h>
#include <hip/hip_bf16.h>

typedef float v2f __attribute__((ext_vector_type(2)));
typedef float v8f __attribute__((ext_vector_type(8)));

#define N_RES 384
#define N_SEQ 256
#define DS    384
#define DP    128
#define NCLS  22
#define GAP   21

// d_out layout: logits [256*384] | eij [384*384*22] | ei [384]
#define OFF_EIJ (N_SEQ * N_RES)                      // 98304
#define OFF_EI  (OFF_EIJ + N_RES * N_RES * NCLS)     // 3342336

// ---------------------------------------------------------------------------
// CDNA5 async global->LDS helpers (GLOBAL_LOAD_ASYNC_TO_LDS_B128, ASYNCcnt)
// ---------------------------------------------------------------------------
typedef __attribute__((address_space(3))) float lds_float;

__device__ __forceinline__ unsigned lds_addr(void* p) {
    // addrspacecast generic -> LDS; AS3 pointers are the 32-bit LDS byte offset
    return (unsigned)(uintptr_t)(lds_float*)p;
}

__device__ __forceinline__ void async_copy_b128(unsigned lds_byte_off, const float* gptr) {
    unsigned long long ga = (unsigned long long)(uintptr_t)gptr;
    asm volatile("global_load_async_to_lds_b128 %0, %1, off"
                 :: "v"(lds_byte_off), "v"(ga) : "memory");
}

__device__ __forceinline__ void wait_async0() {
    asm volatile("s_wait_asynccnt 0" ::: "memory");
}

// ---------------------------------------------------------------------------
// 1) Pack msa int32 -> u8 with gap masked to 0xFF
// ---------------------------------------------------------------------------
__global__ void k_pack_msa(const int* __restrict__ msa, unsigned char* __restrict__ out) {
    int idx = blockIdx.x * blockDim.x + threadIdx.x;
    if (idx < N_SEQ * N_RES) {
        int c = msa[idx];
        out[idx] = (c == GAP) ? 0xFFu : (unsigned char)c;
    }
}

// ---------------------------------------------------------------------------
// 2) h = GELU(si @ W1 + b1), f32 WMMA 16x16x4, one 16x16 tile per wave
// ---------------------------------------------------------------------------
__global__ __launch_bounds__(256) void k_gemm1_gelu(
    const float* __restrict__ si, const float* __restrict__ W1,
    const float* __restrict__ b1, float* __restrict__ h) {
    int t    = threadIdx.x;
    int wave = t >> 5;
    int lane = t & 31;
    int tile = blockIdx.x * 8 + wave;          // 576 tiles total, 8 waves/block
    int tm   = (tile / (DS / 16)) * 16;
    int tn   = (tile % (DS / 16)) * 16;
    int half = lane >> 4;
    int ml   = lane & 15;

    v8f acc = {};
    for (int k0 = 0; k0 < DS; k0 += 4) {
        int ka = k0 + 2 * half;                // ISA 16x4 A layout: lanes16-31 hold K+2,K+3
        v2f a, b;
        a.x = si[(tm + ml) * DS + ka];
        a.y = si[(tm + ml) * DS + ka + 1];
        b.x = W1[ka * DS + tn + ml];           // B 4x16: row K striped across lanes
        b.y = W1[(ka + 1) * DS + tn + ml];
        acc = __builtin_amdgcn_wmma_f32_16x16x4_f32(
            false, a, false, b, (short)0, acc, false, false);
    }
    int col = tn + ml;
    float bias = b1[col];
#pragma unroll
    for (int r = 0; r < 8; ++r) {
        int row = tm + r + 8 * half;           // C/D: VGPR r -> rows r / r+8
        float v = acc[r] + bias;
        float g = 0.5f * v * (1.0f + erff(v * 0.70710678118654752f)); // exact GELU
        h[row * DS + col] = g;
    }
}

// ---------------------------------------------------------------------------
// 3) ei[i] = LayerNorm(h[i,:]) @ W2 + b2   (one block of 128 per row)
// ---------------------------------------------------------------------------
__global__ __launch_bounds__(128) void k_ln_ei(
    const float* __restrict__ h, const float* __restrict__ g,
    const float* __restrict__ bln, const float* __restrict__ W2,
    const float* __restrict__ b2, float* __restrict__ ei) {
    __shared__ float red[128];
    int i = blockIdx.x, t = threadIdx.x;
    float x0 = h[i * DS + t], x1 = h[i * DS + t + 128], x2 = h[i * DS + t + 256];

    red[t] = x0 + x1 + x2;  __syncthreads();
    for (int s = 64; s > 0; s >>= 1) { if (t < s) red[t] += red[t + s]; __syncthreads(); }
    float mu = red[0] / DS; __syncthreads();

    float d0 = x0 - mu, d1 = x1 - mu, d2 = x2 - mu;
    red[t] = d0 * d0 + d1 * d1 + d2 * d2;  __syncthreads();
    for (int s = 64; s > 0; s >>= 1) { if (t < s) red[t] += red[t + s]; __syncthreads(); }
    float rstd = rsqrtf(red[0] / DS + 1e-5f); __syncthreads();

    red[t] = (d0 * rstd * g[t      ] + bln[t      ]) * W2[t      ]
           + (d1 * rstd * g[t + 128] + bln[t + 128]) * W2[t + 128]
           + (d2 * rstd * g[t + 256] + bln[t + 256]) * W2[t + 256];
    __syncthreads();
    for (int s = 64; s > 0; s >>= 1) { if (t < s) red[t] += red[t + s]; __syncthreads(); }
    if (t == 0) ei[i] = red[0] + b2[0];
}

// ---------------------------------------------------------------------------
// 4) eij = (LN(0.5*(z_ij + z_ji)) @ Wp + bp), diagonal zeroed.
//    16 pairs per block (all share row i). z tiles staged into LDS via
//    CDNA5 async DMA (ASYNCcnt); waves 0/1 run f32 WMMA over padded Wp.
// ---------------------------------------------------------------------------
__global__ __launch_bounds__(256) void k_pair_head(
    const float* __restrict__ zij, const float* __restrict__ lg,
    const float* __restrict__ lb, const float* __restrict__ Wp,
    const float* __restrict__ bp, float* __restrict__ eij) {
    __shared__ float zA [16 * DP];             // z[i, j0+p, :]
    __shared__ float zB [16 * DP];             // z[j0+p, i, :]
    __shared__ float aln[16][DP];
    __shared__ float wp [DP][32];              // Wp zero-padded 22 -> 32 cols

    int t  = threadIdx.x;
    int p0 = blockIdx.x * 16;
    int i  = p0 / N_RES;
    int j0 = p0 % N_RES;                       // 384 % 16 == 0 -> whole tile shares i

    // ---- async DMA both z tiles into LDS: 512 x B128 slots per buffer ----
    unsigned baseA = lds_addr(zA);
    unsigned baseB = lds_addr(zB);
#pragma unroll
    for (int it = 0; it < 2; ++it) {
        int slot = t + 256 * it;               // slot covers floats [slot*4, slot*4+4)
        int p = slot >> 5;
        int c = (slot & 31) * 4;
        async_copy_b128(baseA + slot * 16, &zij[((size_t)i * N_RES + (j0 + p)) * DP + c]);
        async_copy_b128(baseB + slot * 16, &zij[((size_t)(j0 + p) * N_RES + i) * DP + c]);
    }
    // ---- padded Wp via regular loads (needs zero-fill) ----
#pragma unroll
    for (int it = 0; it < 16; ++it) {
        int slot = t + 256 * it;
        int k = slot >> 5, n = slot & 31;
        wp[k][n] = (n < NCLS) ? Wp[k * NCLS + n] : 0.0f;
    }
    wait_async0();
    __syncthreads();

    // LayerNorm of zs = 0.5*(zA+zB): 16 lanes per pair, 8 channels per lane
    int p = t >> 4, s = t & 15;
    float x[8]; float sum = 0.0f;
#pragma unroll
    for (int q = 0; q < 8; ++q) {
        int c = s * 8 + q;
        x[q] = 0.5f * (zA[p * DP + c] + zB[p * DP + c]);
        sum += x[q];
    }
    for (int m = 8; m >= 1; m >>= 1) sum += __shfl_xor(sum, m, 32);
    float mu = sum / DP;
    float vs = 0.0f;
#pragma unroll
    for (int q = 0; q < 8; ++q) { float d = x[q] - mu; vs += d * d; }
    for (int m = 8; m >= 1; m >>= 1) vs += __shfl_xor(vs, m, 32);
    float rstd = rsqrtf(vs / DP + 1e-5f);
#pragma unroll
    for (int q = 0; q < 8; ++q) {
        int c = s * 8 + q;
        aln[p][c] = (x[q] - mu) * rstd * lg[c] + lb[c];
    }
    __syncthreads();

    int wave = t >> 5;
    if (wave < 2) {                            // two N-tiles cover 22 (+10 pad) classes
        int lane = t & 31;
        int half = lane >> 4, ml = lane & 15;
        int n = ml + 16 * wave;
        v8f acc = {};
        for (int k0 = 0; k0 < DP; k0 += 4) {
            int ka = k0 + 2 * half;
            v2f a, b;
            a.x = aln[ml][ka];  a.y = aln[ml][ka + 1];
            b.x = wp[ka][n];    b.y = wp[ka + 1][n];
            acc = __builtin_amdgcn_wmma_f32_16x16x4_f32(
                false, a, false, b, (short)0, acc, false, false);
        }
        if (n < NCLS) {
            float bias = bp[n];
#pragma unroll
            for (int r = 0; r < 8; ++r) {
                int pr = r + 8 * half;
                int j  = j0 + pr;
                float v = acc[r] + bias;
                if (j == i) v = 0.0f;          // zero diagonal
                eij[((size_t)i * N_RES + j) * NCLS + n] = v;
            }
        }
    }
}

// ---------------------------------------------------------------------------
// 5) logits[m,i] = ei[i] + sum_j eij[i,j,msa[m,j]] (gap masked)
//    One block per i: eij row (33.8 KB) async-DMA'd to LDS; thread m streams
//    packed msa (L2-resident) 4 classes per u32 load.
// ---------------------------------------------------------------------------
__global__ __launch_bounds__(256) void k_einsum(
    const float* __restrict__ eij, const float* __restrict__ ei,
    const unsigned char* __restrict__ msa8, float* __restrict__ logits) {
    __shared__ float row[N_RES * NCLS];        // 8448 floats = 33792 B (2112 B128 slots)
    int i = blockIdx.x, t = threadIdx.x;
    const float* src = eij + (size_t)i * N_RES * NCLS;
    unsigned base = lds_addr(row);
    for (int slot = t; slot < (N_RES * NCLS) / 4; slot += 256)
        async_copy_b128(base + slot * 16, src + slot * 4);
    wait_async0();
    __syncthreads();

    int m = t;                                 // N_SEQ == blockDim.x == 256
    const unsigned int* mrow = (const unsigned int*)(msa8 + (size_t)m * N_RES);
    float acc = 0.0f;
    for (int j4 = 0; j4 < N_RES / 4; ++j4) {
        unsigned int pk = mrow[j4];
        int j = j4 * 4;
#pragma unroll
        for (int k = 0; k < 4; ++k) {
            unsigned int c = (pk >> (8 * k)) & 0xFFu;
            if (c < NCLS) acc += row[(j + k) * NCLS + c];   // gap (0xFF) contributes 0
        }
    }
    logits[(size_t)m * N_RES + i] = ei[i] + acc;
}

// ---------------------------------------------------------------------------
extern "C" void kernel_launch(void* const* d_in, const int* in_sizes, int n_in,
                              void* d_out, int out_size, void* d_ws, size_t ws_size,
                              hipStream_t stream) {
    const float* si    = (const float*)d_in[0];
    const float* zij   = (const float*)d_in[1];
    const int*   msa   = (const int*)  d_in[2];
    const float* W1    = (const float*)d_in[3];
    const float* b1    = (const float*)d_in[4];
    const float* ln1_g = (const float*)d_in[5];
    const float* ln1_b = (const float*)d_in[6];
    const float* W2    = (const float*)d_in[7];
    const float* b2    = (const float*)d_in[8];
    const float* lnp_g = (const float*)d_in[9];
    const float* lnp_b = (const float*)d_in[10];
    const float* Wp    = (const float*)d_in[11];
    const float* bp    = (const float*)d_in[12];

    float* out    = (float*)d_out;
    float* logits = out;
    float* eij    = out + OFF_EIJ;
    float* ei     = out + OFF_EI;

    float*         h    = (float*)d_ws;                                   // 384*384 f32
    unsigned char* msa8 = (unsigned char*)d_ws + (size_t)N_RES * DS * 4;  // 98304 u8

    hipLaunchKernelGGL(k_pack_msa, dim3((N_SEQ * N_RES + 255) / 256), dim3(256), 0, stream,
                       msa, msa8);
    hipLaunchKernelGGL(k_gemm1_gelu, dim3((DS / 16) * (DS / 16) / 8), dim3(256), 0, stream,
                       si, W1, b1, h);
    hipLaunchKernelGGL(k_ln_ei, dim3(N_RES), dim3(128), 0, stream,
                       h, ln1_g, ln1_b, W2, b2, ei);
    hipLaunchKernelGGL(k_pair_head, dim3(N_RES * N_RES / 16), dim3(256), 0, stream,
                       zij, lnp_g, lnp_b, Wp, bp, eij);
    hipLaunchKernelGGL(k_einsum, dim3(N_RES), dim3(256), 0, stream,
                       eij, ei, msa8, logits);
}